// GCN_2327872274540
// MI455X (gfx1250) — compile-verified
//
#include <hip/hip_runtime.h>
#include <hip/hip_bf16.h>

typedef __attribute__((ext_vector_type(2))) float v2f;
typedef __attribute__((ext_vector_type(8))) float v8f;

#define D 64

// ---------------- degree kernels ----------------
__global__ __launch_bounds__(256) void deg_init(float* dout, float* din, int N) {
    int i = blockIdx.x * blockDim.x + threadIdx.x;
    if (i < N) { dout[i] = 1.0f; din[i] = 1.0f; }   // self-loop contributes 1
}

__global__ __launch_bounds__(256) void deg_count(const int* __restrict__ src,
                                                 const int* __restrict__ dst,
                                                 float* dout, float* din, int E) {
    int e = blockIdx.x * blockDim.x + threadIdx.x;
    if (e < E) {
        atomicAdd(&dout[src[e]], 1.0f);
        atomicAdd(&din[dst[e]], 1.0f);
    }
}

__global__ __launch_bounds__(256) void deg_rsqrt(float* dout, float* din, int N) {
    int i = blockIdx.x * blockDim.x + threadIdx.x;
    if (i < N) { dout[i] = rsqrtf(dout[i]); din[i] = rsqrtf(din[i]); }
}

// ---------------- WMMA GEMM: out = (X * rs[:,None]) @ W, written to out0 and out1 ----------------
// One wave per 16-node tile, 4 column tiles of 16, K=64 in 16 steps of V_WMMA_F32_16X16X4_F32.
__global__ __launch_bounds__(256) void gemm64_wmma(const float* __restrict__ X,
                                                   const float* __restrict__ rs,
                                                   const float* __restrict__ W,
                                                   float* __restrict__ out0,
                                                   float* __restrict__ out1,
                                                   int N) {
    const int wave = threadIdx.x >> 5;
    const int lane = threadIdx.x & 31;
    const int nb = blockIdx.x * 128 + wave * 16;   // wave-uniform guard -> EXEC all-1s in WMMA
    if (nb >= N) return;
    const int half = lane >> 4;                     // K-half selector
    const int lr   = lane & 15;                     // row (A) / col (B,C)

    const float s = rs[nb + lr];
    const float* __restrict__ xrow = X + (size_t)(nb + lr) * D;

    v8f acc0 = {}, acc1 = {}, acc2 = {}, acc3 = {};
#pragma unroll
    for (int k0 = 0; k0 < D; k0 += 4) {
        const int ka = k0 + half * 2;
        v2f a;
        a.x = xrow[ka] * s;
        a.y = xrow[ka + 1] * s;
        const float* __restrict__ w0 = W + (size_t)ka * D + lr;   // row ka
        const float* __restrict__ w1 = w0 + D;                    // row ka+1
        v2f b0 = { w0[0],  w1[0]  };
        v2f b1 = { w0[16], w1[16] };
        v2f b2 = { w0[32], w1[32] };
        v2f b3 = { w0[48], w1[48] };
        acc0 = __builtin_amdgcn_wmma_f32_16x16x4_f32(false, a, false, b0, (short)0, acc0, false, false);
        acc1 = __builtin_amdgcn_wmma_f32_16x16x4_f32(false, a, false, b1, (short)0, acc1, false, false);
        acc2 = __builtin_amdgcn_wmma_f32_16x16x4_f32(false, a, false, b2, (short)0, acc2, false, false);
        acc3 = __builtin_amdgcn_wmma_f32_16x16x4_f32(false, a, false, b3, (short)0, acc3, false, false);
    }

    const int row0 = nb + half * 8;                // C layout: VGPR v -> row half*8+v, col lr
#pragma unroll
    for (int v = 0; v < 8; ++v) {
        const size_t base = (size_t)(row0 + v) * D + lr;
        out0[base]      = acc0[v];
        out0[base + 16] = acc1[v];
        out0[base + 32] = acc2[v];
        out0[base + 48] = acc3[v];
        out1[base]      = acc0[v];
        out1[base + 16] = acc1[v];
        out1[base + 32] = acc2[v];
        out1[base + 48] = acc3[v];
    }
}

// ---------------- edge scatter-add: agg[dst] += h[src]; 16 lanes/edge, float4 ----------------
__global__ __launch_bounds__(256) void scatter_add(const float* __restrict__ h,
                                                   const int* __restrict__ src,
                                                   const int* __restrict__ dst,
                                                   float* __restrict__ agg, int E) {
    const int t = blockIdx.x * blockDim.x + threadIdx.x;
    const int e = t >> 4;
    if (e >= E) return;
    const int f = (t & 15) * 4;
    const int s = src[e];
    const int d = dst[e];
    const float4 v = *(const float4*)(h + (size_t)s * D + f);
    float* p = agg + (size_t)d * D + f;
    atomicAdd(p + 0, v.x);
    atomicAdd(p + 1, v.y);
    atomicAdd(p + 2, v.z);
    atomicAdd(p + 3, v.w);
}

// ---------------- post: hout = relu(agg * rs_in + b) ----------------
__global__ __launch_bounds__(256) void post_scale_bias(const float* __restrict__ agg,
                                                       const float* __restrict__ rs_in,
                                                       const float* __restrict__ b,
                                                       float* __restrict__ hout, int N) {
    const int t = blockIdx.x * blockDim.x + threadIdx.x;
    const int node = t >> 4;
    if (node >= N) return;
    const int f = (t & 15) * 4;
    const float s = rs_in[node];
    const float4 v = *(const float4*)(agg + (size_t)node * D + f);
    float4 r;
    r.x = fmaxf(v.x * s + b[f + 0], 0.0f);
    r.y = fmaxf(v.y * s + b[f + 1], 0.0f);
    r.z = fmaxf(v.z * s + b[f + 2], 0.0f);
    r.w = fmaxf(v.w * s + b[f + 3], 0.0f);
    *(float4*)(hout + (size_t)node * D + f) = r;
}

// ---------------- final: out[n] = dot(agg[n]*rs_in[n] + b2, Wr) + br; one wave per node ----------------
__global__ __launch_bounds__(256) void final_dot(const float* __restrict__ agg,
                                                 const float* __restrict__ rs_in,
                                                 const float* __restrict__ b2,
                                                 const float* __restrict__ Wr,
                                                 const float* __restrict__ br,
                                                 float* __restrict__ out, int N) {
    const int wave = threadIdx.x >> 5;
    const int lane = threadIdx.x & 31;
    const int node = blockIdx.x * 8 + wave;
    if (node >= N) return;
    const float s = rs_in[node];
    const float* row = agg + (size_t)node * D;
    float acc = (row[lane]      * s + b2[lane])      * Wr[lane]
              + (row[lane + 32] * s + b2[lane + 32]) * Wr[lane + 32];
#pragma unroll
    for (int off = 16; off > 0; off >>= 1) acc += __shfl_xor(acc, off, 32);
    if (lane == 0) out[node] = acc + br[0];
}

extern "C" void kernel_launch(void* const* d_in, const int* in_sizes, int n_in,
                              void* d_out, int out_size, void* d_ws, size_t ws_size,
                              hipStream_t stream) {
    const float* x   = (const float*)d_in[0];
    const int*   src = (const int*)d_in[1];
    const int*   dst = (const int*)d_in[2];
    const float* W1  = (const float*)d_in[3];
    const float* b1  = (const float*)d_in[4];
    const float* W2  = (const float*)d_in[5];
    const float* b2  = (const float*)d_in[6];
    const float* Wr  = (const float*)d_in[7];
    const float* br  = (const float*)d_in[8];
    float* out = (float*)d_out;

    const int N = in_sizes[0] / D;
    const int E = in_sizes[1];

    float* ws     = (float*)d_ws;
    float* rs_out = ws;                      // [N]
    float* rs_in  = ws + N;                  // [N]
    float* h      = ws + 2 * (size_t)N;      // [N,64]
    float* agg    = h + (size_t)N * D;       // [N,64]

    const int TB = 256;
    const int gN    = (N + TB - 1) / TB;
    const int gE    = (E + TB - 1) / TB;
    const int gGemm = (N + 127) / 128;
    const int gEdge = (int)(((size_t)E * 16 + TB - 1) / TB);
    const int gFeat = (int)(((size_t)N * 16 + TB - 1) / TB);
    const int gNode = (N + 7) / 8;

    // degrees (with self loops; clip(.,1) is a no-op since deg >= 1)
    deg_init <<<gN, TB, 0, stream>>>(rs_out, rs_in, N);
    deg_count<<<gE, TB, 0, stream>>>(src, dst, rs_out, rs_in, E);
    deg_rsqrt<<<gN, TB, 0, stream>>>(rs_out, rs_in, N);

    // layer 1: h = (x*rs_out)@W1 ; agg starts as self-loop term h
    gemm64_wmma<<<gGemm, TB, 0, stream>>>(x, rs_out, W1, h, agg, N);
    scatter_add<<<gEdge, TB, 0, stream>>>(h, src, dst, agg, E);
    post_scale_bias<<<gFeat, TB, 0, stream>>>(agg, rs_in, b1, h, N);   // relu(agg*rs_in + b1) -> h

    // layer 2 (GEMM in-place on h: each wave reads only its own 16 rows before storing them)
    gemm64_wmma<<<gGemm, TB, 0, stream>>>(h, rs_out, W2, h, agg, N);
    scatter_add<<<gEdge, TB, 0, stream>>>(h, src, dst, agg, E);

    // regressor head
    final_dot<<<gNode, TB, 0, stream>>>(agg, rs_in, b2, Wr, br, out, N);
}